// InteractionArch_15324443312162
// MI455X (gfx1250) — compile-verified
//
#include <hip/hip_runtime.h>

typedef __attribute__((ext_vector_type(2))) float v2f;
typedef __attribute__((ext_vector_type(8))) float v8f;

#define BATCH       16384
#define NUM_SPARSE  26
#define FTOT        27            // 1 dense row + 26 sparse rows
#define EMBED       128
#define NPAIR       ((FTOT * (FTOT - 1)) / 2)   // 351
#define OUT_COLS    (EMBED + NPAIR)             // 479
#define CHUNK       32            // K columns staged per LDS pass
#define LROW        34            // padded LDS row stride in floats (bank-conflict-free)
#define WPB         8             // waves (batch elements) per block

__global__ __launch_bounds__(256)
void dlrm_interaction_wmma(const float* __restrict__ dense,
                           const float* __restrict__ sparse,
                           float* __restrict__ out)
{
    __shared__ float shm[WPB * 32 * LROW];   // ~34 KB

    const int tid  = threadIdx.x;
    const int wave = tid >> 5;
    const int lane = tid & 31;
    const int b    = blockIdx.x * WPB + wave;

    float* smw = &shm[wave * 32 * LROW];

    const int h   = lane >> 4;   // half-wave select (K sub-pair / M+8)
    const int n16 = lane & 15;   // row/col within 16

    v8f c00 = {}; v8f c01 = {}; v8f c11 = {};

    const float* drow = dense  + (size_t)b * EMBED;
    const float* srow = sparse + (size_t)b * NUM_SPARSE * EMBED;

    // staging lane mapping: 4 rows x 32 cols (float4 per lane) per iteration
    const int srо_row = lane >> 3;        // 0..3
    const int srо_col = (lane & 7) * 4;   // 0,4,...,28

    for (int ch = 0; ch < EMBED / CHUNK; ++ch) {
        const int cbase = ch * CHUNK;

        // ---- stage 32 rows x CHUNK cols into LDS (rows >= 27 zeroed) ----
        #pragma unroll
        for (int it = 0; it < 8; ++it) {
            const int r = it * 4 + srо_row;
            float x0 = 0.f, x1 = 0.f, x2 = 0.f, x3 = 0.f;
            if (r < FTOT) {
                const float* p = (r == 0)
                    ? (drow + cbase + srо_col)
                    : (srow + (size_t)(r - 1) * EMBED + cbase + srо_col);
                x0 = p[0]; x1 = p[1]; x2 = p[2]; x3 = p[3];
            }
            float* q = &smw[r * LROW + srо_col];
            *(v2f*)(q)     = (v2f){x0, x1};
            *(v2f*)(q + 2) = (v2f){x2, x3};
        }
        __syncthreads();

        // ---- K loop: 8 steps of K=4, 3 WMMA tiles each ----
        #pragma unroll
        for (int s = 0; s < CHUNK / 4; ++s) {
            const int col = s * 4 + 2 * h;
            v2f a0 = *(const v2f*)&smw[n16 * LROW + col];         // rows 0..15
            v2f a1 = *(const v2f*)&smw[(16 + n16) * LROW + col];  // rows 16..31
            c00 = __builtin_amdgcn_wmma_f32_16x16x4_f32(false, a0, false, a0,
                                                        (short)0, c00, false, false);
            c01 = __builtin_amdgcn_wmma_f32_16x16x4_f32(false, a0, false, a1,
                                                        (short)0, c01, false, false);
            c11 = __builtin_amdgcn_wmma_f32_16x16x4_f32(false, a1, false, a1,
                                                        (short)0, c11, false, false);
        }
        __syncthreads();
    }

    // ---- epilogue ----
    float* orow = out + (size_t)b * OUT_COLS;

    // dense passthrough (coalesced)
    #pragma unroll
    for (int t = 0; t < 4; ++t)
        orow[lane + 32 * t] = drow[lane + 32 * t];

    // strict upper triangle scatter: off(i,j) = i*(53-i)/2 + j-i-1
    float* ibase = orow + EMBED;
    #pragma unroll
    for (int r = 0; r < 8; ++r) {
        const int i0 = r + 8 * h;        // M within tile
        // tile (0,0): i=i0, j=n16
        if (i0 < n16)
            ibase[(i0 * (53 - i0)) / 2 + (n16 - i0 - 1)] = c00[r];
        // tile (0,1): i=i0 (<16), j=16+n16  -> i<j always
        {
            const int j = 16 + n16;
            if (j < FTOT)
                ibase[(i0 * (53 - i0)) / 2 + (j - i0 - 1)] = c01[r];
        }
        // tile (1,1): i=16+i0, j=16+n16
        {
            const int i = 16 + i0, j = 16 + n16;
            if (i < j && j < FTOT)
                ibase[(i * (53 - i)) / 2 + (j - i - 1)] = c11[r];
        }
    }
}

extern "C" void kernel_launch(void* const* d_in, const int* in_sizes, int n_in,
                              void* d_out, int out_size, void* d_ws, size_t ws_size,
                              hipStream_t stream)
{
    const float* dense  = (const float*)d_in[0];   // (16384, 128) fp32
    const float* sparse = (const float*)d_in[1];   // (16384, 26, 128) fp32
    float* out = (float*)d_out;                    // (16384, 479) fp32

    dim3 grid(BATCH / WPB);   // 2048 blocks, 8 waves each, 1 batch element per wave
    dim3 block(256);
    hipLaunchKernelGGL(dlrm_interaction_wmma, grid, block, 0, stream,
                       dense, sparse, out);
}